// RandLANet_70901320122518
// MI455X (gfx1250) — compile-verified
//
#include <hip/hip_runtime.h>

typedef __attribute__((ext_vector_type(2))) float v2f;
typedef __attribute__((ext_vector_type(8))) float v8f;

static inline int ceildiv_l(long a, long b) { return (int)((a + b - 1) / b); }

// ---------------------------------------------------------------------------
// GEMM: Y[b][o][p] = bias[o] + sum_c W[o][c] * X[b][c][p]
// Channels-first layout, explicit batch strides (xbs/ybs) so we can write
// directly into sub-channel-ranges of larger concat buffers.
// ---------------------------------------------------------------------------

// WMMA fp32 path: one wave32 per 16x32 output tile (two 16x16 accumulators),
// K stepped by 4, two back-to-back v_wmma per step. Requires P % 32 == 0
// (guaranteed by the dispatcher), so no bounds checks / exec-mask churn.
__global__ __launch_bounds__(32) void k_gemm_wmma(
    const float* __restrict__ W, const float* __restrict__ bias,
    const float* __restrict__ X, long xbs,
    float* __restrict__ Y, long ybs, int Cin, int P)
{
  int lane = threadIdx.x & 31;
  int half = lane >> 4;
  int l = lane & 15;
  int p0 = blockIdx.x * 32;
  int o0 = blockIdx.y * 16;
  int b  = blockIdx.z;
  const float* Xb = X + (long)b * xbs;
  float* Yb = Y + (long)b * ybs;
  int col0 = p0 + l;
  int col1 = p0 + 16 + l;

  v8f acc0 = {};
  v8f acc1 = {};
  for (int k = 0; k < Cin; k += 4) {
    int ka = k + 2 * half;
    // A fragment (16x4 of W): lanes 0-15 -> row o0+l, Ks {k,k+1}; lanes 16-31 Ks {k+2,k+3}
    const float* wrow = W + (long)(o0 + l) * Cin + ka;
    v2f a;
    a[0] = wrow[0];
    a[1] = wrow[1];
    // B fragments (4x16 of X): vgpr v holds rows K=v (lanes 0-15) / K=v+2 (lanes 16-31)
    const float* xk = Xb + (long)ka * P;
    v2f b0, b1;
    b0[0] = xk[col0];
    b0[1] = xk[P + col0];
    b1[0] = xk[col1];
    b1[1] = xk[P + col1];
    acc0 = __builtin_amdgcn_wmma_f32_16x16x4_f32(false, a, false, b0,
                                                 (short)0, acc0, false, false);
    acc1 = __builtin_amdgcn_wmma_f32_16x16x4_f32(false, a, false, b1,
                                                 (short)0, acc1, false, false);
  }
#pragma unroll
  for (int v = 0; v < 8; ++v) {
    int row = o0 + v + (half ? 8 : 0);
    float bi = bias[row];
    Yb[(long)row * P + col0] = acc0[v] + bi;
    Yb[(long)row * P + col1] = acc1[v] + bi;
  }
}

// Scalar fallback for odd channel counts (Cin=10 rel-MLP, layer-1 8-ch MLPs).
__global__ void k_gemm_scalar(
    const float* __restrict__ W, const float* __restrict__ bias,
    const float* __restrict__ X, long xbs,
    float* __restrict__ Y, long ybs, int Cin, int P)
{
  long t = (long)blockIdx.x * 256 + threadIdx.x;
  int o = blockIdx.y;
  int b = blockIdx.z;
  if (t >= P) return;
  const float* Xb = X + (long)b * xbs;
  const float* wr = W + (long)o * Cin;
  float s = bias[o];
  for (int c = 0; c < Cin; ++c) s += wr[c] * Xb[(long)c * P + t];
  Y[(long)b * ybs + (long)o * P + t] = s;
}

// ---------------------------------------------------------------------------
// BatchNorm: per-channel mean/var over (B, P), then apply + leaky-relu.
// ---------------------------------------------------------------------------
__global__ void k_bn_stats(const float* __restrict__ X, long xbs, int P, int B,
                           float* __restrict__ mean, float* __restrict__ rstd)
{
  int c = blockIdx.x;
  __shared__ float ss[256], sq[256];
  float s = 0.0f, q = 0.0f;
  for (int b = 0; b < B; ++b) {
    const float* p = X + (long)b * xbs + (long)c * P;
    for (int i = threadIdx.x; i < P; i += 256) {
      float v = p[i];
      s += v;
      q += v * v;
    }
  }
  ss[threadIdx.x] = s;
  sq[threadIdx.x] = q;
  __syncthreads();
  for (int st = 128; st > 0; st >>= 1) {
    if ((int)threadIdx.x < st) {
      ss[threadIdx.x] += ss[threadIdx.x + st];
      sq[threadIdx.x] += sq[threadIdx.x + st];
    }
    __syncthreads();
  }
  if (threadIdx.x == 0) {
    float cnt = (float)((long)B * (long)P);
    float m = ss[0] / cnt;
    float var = fmaxf(sq[0] / cnt - m * m, 0.0f);
    mean[c] = m;
    rstd[c] = rsqrtf(var + 1e-6f);
  }
}

__global__ void k_bn_apply(float* __restrict__ Y, long ybs, int P,
                           const float* __restrict__ g, const float* __restrict__ beta,
                           const float* __restrict__ mean, const float* __restrict__ rstd,
                           float slope)
{
  long t = (long)blockIdx.x * 256 + threadIdx.x;
  int c = blockIdx.y;
  int b = blockIdx.z;
  if (t >= P) return;
  float* p = Y + (long)b * ybs + (long)c * P + t;
  float v = (*p - mean[c]) * rstd[c] * g[c] + beta[c];
  *p = (v >= 0.0f) ? v : slope * v;
}

// ---------------------------------------------------------------------------
// fc0: y[b][o][n] = sum_d points[b][n][d] * W[o][d] + b[o]   (8 channels)
// ---------------------------------------------------------------------------
__global__ void k_fc0(const float* __restrict__ pts, const float* __restrict__ W,
                      const float* __restrict__ bvec, float* __restrict__ Y, int N)
{
  long t = (long)blockIdx.x * 256 + threadIdx.x;
  int b = blockIdx.z;
  if (t >= N) return;
  const float* p = pts + ((long)b * N + t) * 3;
  float x = p[0], y = p[1], z = p[2];
#pragma unroll
  for (int o = 0; o < 8; ++o) {
    Y[((long)b * 8 + o) * N + t] = W[o * 3 + 0] * x + W[o * 3 + 1] * y +
                                   W[o * 3 + 2] * z + bvec[o];
  }
}

// Apply permutation: coords[b][t][:] = points[b][perm[t]][:]; Xout[b][c][t] = Xin[b][c][perm[t]]
__global__ void k_perm(const float* __restrict__ pts, const float* __restrict__ Xin,
                       const int* __restrict__ perm, float* __restrict__ coords,
                       float* __restrict__ Xout, int N, int C)
{
  long t = (long)blockIdx.x * 256 + threadIdx.x;
  int b = blockIdx.z;
  if (t >= N) return;
  int j = perm[t];
  const float* src = pts + ((long)b * N + j) * 3;
  float* dst = coords + ((long)b * N + t) * 3;
  dst[0] = src[0];
  dst[1] = src[1];
  dst[2] = src[2];
  for (int c = 0; c < C; ++c)
    Xout[((long)b * C + c) * N + t] = Xin[((long)b * C + c) * N + j];
}

// ---------------------------------------------------------------------------
// KNN: per row, top-4 LARGEST sqrt(max(|a|^2+|b|^2-2ab,0)) (matches top_k on dist;
// strict > insertion reproduces tie-break toward lower index). Candidates staged
// through LDS in tiles of 256.
// ---------------------------------------------------------------------------
__global__ void k_knn(const float* __restrict__ coords, int Nfull, int n,
                      int* __restrict__ idx)
{
  int b = blockIdx.y;
  int i = blockIdx.x * 256 + threadIdx.x;
  const float* cb = coords + (long)b * Nfull * 3;
  bool act = (i < n);
  float xi = 0.f, yi = 0.f, zi = 0.f, sqi = 0.f;
  if (act) {
    xi = cb[i * 3 + 0];
    yi = cb[i * 3 + 1];
    zi = cb[i * 3 + 2];
    sqi = xi * xi + yi * yi + zi * zi;
  }
  float bv0 = -1e30f, bv1 = -1e30f, bv2 = -1e30f, bv3 = -1e30f;
  int bi0 = 0, bi1 = 0, bi2 = 0, bi3 = 0;
  __shared__ float tx[256], ty[256], tz[256], tq[256];
  for (int j0 = 0; j0 < n; j0 += 256) {
    int j = j0 + threadIdx.x;
    if (j < n) {
      float a = cb[j * 3 + 0], c1 = cb[j * 3 + 1], c2 = cb[j * 3 + 2];
      tx[threadIdx.x] = a;
      ty[threadIdx.x] = c1;
      tz[threadIdx.x] = c2;
      tq[threadIdx.x] = a * a + c1 * c1 + c2 * c2;
    }
    __syncthreads();
    int lim = min(256, n - j0);
    if (act) {
      for (int jj = 0; jj < lim; ++jj) {
        float d2 = sqi + tq[jj] - 2.0f * (xi * tx[jj] + yi * ty[jj] + zi * tz[jj]);
        float v = sqrtf(fmaxf(d2, 0.0f));
        int jg = j0 + jj;
        if (v > bv0) {
          bv3 = bv2; bi3 = bi2; bv2 = bv1; bi2 = bi1; bv1 = bv0; bi1 = bi0;
          bv0 = v; bi0 = jg;
        } else if (v > bv1) {
          bv3 = bv2; bi3 = bi2; bv2 = bv1; bi2 = bi1;
          bv1 = v; bi1 = jg;
        } else if (v > bv2) {
          bv3 = bv2; bi3 = bi2;
          bv2 = v; bi2 = jg;
        } else if (v > bv3) {
          bv3 = v; bi3 = jg;
        }
      }
    }
    __syncthreads();
  }
  if (act) {
    int* o = idx + ((long)b * n + i) * 4;
    o[0] = bi0; o[1] = bi1; o[2] = bi2; o[3] = bi3;
  }
}

// rel features (B,10,n,4): [rd, rp(3), ext(3), nbc(3)]
__global__ void k_rel(const float* __restrict__ coords, int Nfull, int n,
                      const int* __restrict__ idx, float* __restrict__ rel)
{
  long t = (long)blockIdx.x * 256 + threadIdx.x;  // over n*4
  int b = blockIdx.z;
  long P = (long)n * 4;
  if (t >= P) return;
  long m = t >> 2;
  int kk = (int)(t & 3);
  const float* cb = coords + (long)b * Nfull * 3;
  int j = idx[((long)b * n + m) * 4 + kk];
  float ex = cb[m * 3 + 0], ey = cb[m * 3 + 1], ez = cb[m * 3 + 2];
  float nx = cb[j * 3 + 0], ny = cb[j * 3 + 1], nz = cb[j * 3 + 2];
  float rx = ex - nx, ry = ey - ny, rz = ez - nz;
  float rd = sqrtf(rx * rx + ry * ry + rz * rz);
  float* rb = rel + (long)b * 10 * P + t;
  rb[0 * P] = rd;
  rb[1 * P] = rx; rb[2 * P] = ry; rb[3 * P] = rz;
  rb[4 * P] = ex; rb[5 * P] = ey; rb[6 * P] = ez;
  rb[7 * P] = nx; rb[8 * P] = ny; rb[9 * P] = nz;
}

// gather: dst[b][c][m][kk] = src[b][c][idx[b][m][kk]]
__global__ void k_gather(const float* __restrict__ src, long sbs,
                         float* __restrict__ dst, long dbs,
                         const int* __restrict__ idx, int n, int C)
{
  long t = (long)blockIdx.x * 256 + threadIdx.x;
  int b = blockIdx.z;
  long P = (long)n * 4;
  long tot = (long)C * P;
  if (t >= tot) return;
  long c = t / P;
  long r = t - c * P;
  long m = r >> 2;
  int kk = (int)(r & 3);
  int j = idx[((long)b * n + m) * 4 + kk];
  dst[(long)b * dbs + c * P + r] = src[(long)b * sbs + c * n + j];
}

// attentive pool: softmax over k=4 of S, weighted sum of X -> F (B,C,n)
__global__ void k_attpool(const float* __restrict__ S, const float* __restrict__ X,
                          long bs4, float* __restrict__ F, long bsF, int n, int C)
{
  long t = (long)blockIdx.x * 256 + threadIdx.x;  // over C*n
  int b = blockIdx.z;
  long tot = (long)C * n;
  if (t >= tot) return;
  long c = t / n;
  long m = t - c * n;
  const float* sp = S + (long)b * bs4 + c * (long)n * 4 + m * 4;
  const float* xp = X + (long)b * bs4 + c * (long)n * 4 + m * 4;
  float mx = fmaxf(fmaxf(sp[0], sp[1]), fmaxf(sp[2], sp[3]));
  float e0 = expf(sp[0] - mx), e1 = expf(sp[1] - mx);
  float e2 = expf(sp[2] - mx), e3 = expf(sp[3] - mx);
  float se = e0 + e1 + e2 + e3;
  F[(long)b * bsF + t] = (e0 * xp[0] + e1 * xp[1] + e2 * xp[2] + e3 * xp[3]) / se;
}

__global__ void k_add_lrelu(const float* __restrict__ A, const float* __restrict__ Bv,
                            float* __restrict__ O, long total, float slope)
{
  long t = (long)blockIdx.x * 256 + threadIdx.x;
  if (t >= total) return;
  float v = A[t] + Bv[t];
  O[t] = (v >= 0.0f) ? v : slope * v;
}

// decimation: dst (B,C,nOut) = src (B,C,nIn)[:, :, :nOut]
__global__ void k_slice(const float* __restrict__ src, long sbs,
                        float* __restrict__ dst, long dbs, int nOut, int nIn, int C)
{
  long t = (long)blockIdx.x * 256 + threadIdx.x;
  int b = blockIdx.z;
  long tot = (long)C * nOut;
  if (t >= tot) return;
  long c = t / nOut;
  long m = t - c * nOut;
  dst[(long)b * dbs + c * nOut + m] = src[(long)b * sbs + c * nIn + m];
}

// ---------------------------------------------------------------------------
// Host-side orchestration
// ---------------------------------------------------------------------------
static void gemm(const float* W, const float* bias, const float* X, long xbs,
                 float* Y, long ybs, int Cin, int Cout, int P, int B, hipStream_t s)
{
  if ((Cout % 16) == 0 && (Cin % 4) == 0 && (P % 32) == 0) {
    dim3 g(P / 32, Cout / 16, B);
    k_gemm_wmma<<<g, dim3(32), 0, s>>>(W, bias, X, xbs, Y, ybs, Cin, P);
  } else {
    dim3 g(ceildiv_l(P, 256), Cout, B);
    k_gemm_scalar<<<g, dim3(256), 0, s>>>(W, bias, X, xbs, Y, ybs, Cin, P);
  }
}

static void bn(float* Y, long ybs, int C, int P, int B, const float* g,
               const float* beta, float* stat, float slope, hipStream_t s)
{
  k_bn_stats<<<dim3(C), dim3(256), 0, s>>>(Y, ybs, P, B, stat, stat + 512);
  dim3 gr(ceildiv_l(P, 256), C, B);
  k_bn_apply<<<gr, dim3(256), 0, s>>>(Y, ybs, P, g, beta, stat, stat + 512, slope);
}

// workspace layout (float offsets)
static const long OFF_COORDS = 0;        //  49152 : coords (B,N,3) permuted
static const long OFF_FIN    = 49152;    // 131072 : layer input feats (B,C,n)
static const long OFF_X1     = 180224;   // 131072 : mlp1 output (B,h,n)
static const long OFF_REL    = 311296;   // 655360 : rel feats (B,10,n,4)
static const long OFF_CAT    = 966656;   // 1048576: concat1 (B,dout,n,4)  [upper half = rf1]
static const long OFF_CAT2   = 2015232;  // 1048576: concat2 (B,dout,n,4)
static const long OFF_S      = 3063808;  // 1048576: attention scores
static const long OFF_POOLF  = 4112384;  // 262144 : pooled feats (B,dout,n)
static const long OFF_P1     = 4374528;  // 131072 : pool1 mlp out (B,h,n)
static const long OFF_P2     = 4505600;  // 262144 : pool2 mlp out (B,dout,n)
static const long OFF_M2     = 4767744;  // 524288 : mlp2 out (B,2dout,n)
static const long OFF_SH     = 5292032;  // 524288 : shortcut out (B,2dout,n)
static const long OFF_STAT   = 5816320;  //   1024 : mean[512] + rstd[512]
static const long OFF_TMP0   = 5817344;  // 131072 : fc0 pre-permute feats
static const long OFF_IDX    = 5948416;  //  65536 ints : knn indices

extern "C" void kernel_launch(void* const* d_in, const int* in_sizes, int n_in,
                              void* d_out, int out_size, void* d_ws, size_t ws_size,
                              hipStream_t stream)
{
  (void)in_sizes; (void)n_in; (void)out_size; (void)ws_size;
  const int B = 2, N = 8192;
  float* ws = (float*)d_ws;

  const float* points = (const float*)d_in[0];
  const int*   perm   = (const int*)d_in[1];
  const float* bn0b   = (const float*)d_in[2];
  const float* bn0g   = (const float*)d_in[3];
  const float* fc0W   = (const float*)d_in[4];
  const float* fc0b   = (const float*)d_in[5];

  float* COORDS = ws + OFF_COORDS;
  float* FIN    = ws + OFF_FIN;
  float* X1     = ws + OFF_X1;
  float* REL    = ws + OFF_REL;
  float* CAT    = ws + OFF_CAT;
  float* CAT2   = ws + OFF_CAT2;
  float* S      = ws + OFF_S;
  float* POOLF  = ws + OFF_POOLF;
  float* P1     = ws + OFF_P1;
  float* P2     = ws + OFF_P2;
  float* M2     = ws + OFF_M2;
  float* SH     = ws + OFF_SH;
  float* STAT   = ws + OFF_STAT;
  float* TMP0   = ws + OFF_TMP0;
  int*   IDX    = (int*)(ws + OFF_IDX);

  // stage 0: fc0 -> BN -> lrelu(0.2) -> permute coords + feats
  k_fc0<<<dim3(ceildiv_l(N, 256), 1, B), dim3(256), 0, stream>>>(points, fc0W, fc0b, TMP0, N);
  bn(TMP0, 8L * N, 8, N, B, bn0g, bn0b, STAT, 0.2f, stream);
  k_perm<<<dim3(ceildiv_l(N, 256), 1, B), dim3(256), 0, stream>>>(points, TMP0, perm,
                                                                  COORDS, FIN, N, 8);

  const int dims_in[4]  = {8, 32, 128, 256};
  const int dims_out[4] = {16, 64, 128, 256};
  int n = N;

  for (int L = 0; L < 4; ++L) {
    const float* const* pp = (const float* const*)(d_in + 6 + 32 * L);
    // pp: [0..3] lse1 W,b,beta,g | [4..7] lse2 | [8..11] mlp1 | [12..15] mlp2
    //     [16..21] pool1 Ws,bs,mlpW,mlpb,mlpbeta,mlpg | [22..27] pool2 | [28..31] short
    int din = dims_in[L], dout = dims_out[L], h = dout / 2;
    long P4 = (long)n * 4;
    long catbs = (long)dout * P4;

    // KNN (top-4 farthest, matching reference's top_k on distance)
    k_knn<<<dim3(ceildiv_l(n, 256), B), dim3(256), 0, stream>>>(COORDS, N, n, IDX);

    // mlp1: din -> h, BN, lrelu 0.2
    gemm(pp[8], pp[9], FIN, (long)din * n, X1, (long)h * n, din, h, n, B, stream);
    bn(X1, (long)h * n, h, n, B, pp[11], pp[10], STAT, 0.2f, stream);

    // rel features
    k_rel<<<dim3(ceildiv_l(P4, 256), 1, B), dim3(256), 0, stream>>>(COORDS, N, n, IDX, REL);

    // lse1: rf1 = smlp(rel) -> upper half of CAT; gathered mlp1 feats -> lower half
    float* CATrf = CAT + (long)h * P4;
    gemm(pp[0], pp[1], REL, 10 * P4, CATrf, catbs, 10, h, (int)P4, B, stream);
    bn(CATrf, catbs, h, (int)P4, B, pp[3], pp[2], STAT, 0.2f, stream);
    k_gather<<<dim3(ceildiv_l((long)h * P4, 256), 1, B), dim3(256), 0, stream>>>(
        X1, (long)h * n, CAT, catbs, IDX, n, h);

    // pool1: scores -> softmax-weighted sum -> mlp(dout->h) + BN + lrelu
    gemm(pp[16], pp[17], CAT, catbs, S, catbs, dout, dout, (int)P4, B, stream);
    k_attpool<<<dim3(ceildiv_l((long)dout * n, 256), 1, B), dim3(256), 0, stream>>>(
        S, CAT, catbs, POOLF, (long)dout * n, n, dout);
    gemm(pp[18], pp[19], POOLF, (long)dout * n, P1, (long)h * n, dout, h, n, B, stream);
    bn(P1, (long)h * n, h, n, B, pp[21], pp[20], STAT, 0.2f, stream);

    // lse2: rf2 = smlp(rf1) -> upper half of CAT2; gathered pool1 feats -> lower half
    float* CAT2rf = CAT2 + (long)h * P4;
    gemm(pp[4], pp[5], CATrf, catbs, CAT2rf, catbs, h, h, (int)P4, B, stream);
    bn(CAT2rf, catbs, h, (int)P4, B, pp[7], pp[6], STAT, 0.2f, stream);
    k_gather<<<dim3(ceildiv_l((long)h * P4, 256), 1, B), dim3(256), 0, stream>>>(
        P1, (long)h * n, CAT2, catbs, IDX, n, h);

    // pool2: scores -> softmax pool -> mlp(dout->dout) + BN + lrelu
    gemm(pp[22], pp[23], CAT2, catbs, S, catbs, dout, dout, (int)P4, B, stream);
    k_attpool<<<dim3(ceildiv_l((long)dout * n, 256), 1, B), dim3(256), 0, stream>>>(
        S, CAT2, catbs, POOLF, (long)dout * n, n, dout);
    gemm(pp[24], pp[25], POOLF, (long)dout * n, P2, (long)dout * n, dout, dout, n, B, stream);
    bn(P2, (long)dout * n, dout, n, B, pp[27], pp[26], STAT, 0.2f, stream);

    // mlp2 (no lrelu) + shortcut (no lrelu), residual add with lrelu 0.01
    gemm(pp[12], pp[13], P2, (long)dout * n, M2, (long)2 * dout * n, dout, 2 * dout, n, B, stream);
    bn(M2, (long)2 * dout * n, 2 * dout, n, B, pp[15], pp[14], STAT, 1.0f, stream);
    gemm(pp[28], pp[29], FIN, (long)din * n, SH, (long)2 * dout * n, din, 2 * dout, n, B, stream);
    bn(SH, (long)2 * dout * n, 2 * dout, n, B, pp[31], pp[30], STAT, 1.0f, stream);
    long tot = (long)B * 2 * dout * n;
    k_add_lrelu<<<dim3(ceildiv_l(tot, 256)), dim3(256), 0, stream>>>(M2, SH, M2, tot, 0.01f);

    // decimate by 4 into FIN for next layer
    int n2 = n / 4;
    k_slice<<<dim3(ceildiv_l((long)2 * dout * n2, 256), 1, B), dim3(256), 0, stream>>>(
        M2, (long)2 * dout * n, FIN, (long)2 * dout * n2, n2, n, 2 * dout);
    n = n2;
  }

  // final smlp: 512 -> 512, BN, lrelu 0.2, straight into d_out (B,512,32,1)
  const float* fW    = (const float*)d_in[134];
  const float* fb    = (const float*)d_in[135];
  const float* fbeta = (const float*)d_in[136];
  const float* fg    = (const float*)d_in[137];
  float* OUT = (float*)d_out;
  gemm(fW, fb, FIN, 512L * n, OUT, 512L * n, 512, 512, n, B, stream);
  bn(OUT, 512L * n, 512, n, B, fg, fbeta, STAT, 0.2f, stream);
}